// GATS_90271622627429
// MI455X (gfx1250) — compile-verified
//
#include <hip/hip_runtime.h>
#include <hip/hip_bf16.h>

typedef __attribute__((ext_vector_type(2))) float v2f;
typedef __attribute__((ext_vector_type(8))) float v8f;

#define GAT_H 4
#define GAT_C 64
#define GAT_HC 256  // H*C
#define LN_EPS_F 1e-5f

// ---- monotonic float<->uint encoding for atomicMax on signed floats ----
__device__ __forceinline__ unsigned enc_f32(float f) {
  unsigned u = __float_as_uint(f);
  return (u & 0x80000000u) ? ~u : (u | 0x80000000u);
}
__device__ __forceinline__ float dec_f32(unsigned u) {
  unsigned v = (u & 0x80000000u) ? (u ^ 0x80000000u) : ~u;
  return __uint_as_float(v);
}

// ======================================================================
// FP32 WMMA GEMM: Hout[N,C] = X[N,K] @ W[K,C]
// One wave per 16x16 output tile; K/4 chained v_wmma_f32_16x16x4_f32.
// A (16x4): lanes0-15 M=0..15 {K=k0,k0+1}, lanes16-31 M=0..15 {K=k0+2,k0+3}
// B (4x16): VGPR = K%2, lane-half = K/2, lanes hold N=0..15
// C/D (16x16): VGPR r -> rows r (lanes0-15) and r+8 (lanes16-31)
// ======================================================================
template <int K>
__global__ void gat_gemm_wmma(const float* __restrict__ X,
                              const float* __restrict__ W,
                              float* __restrict__ Hout,
                              int Nrows, int C) {
  int wid  = (blockIdx.x * blockDim.x + threadIdx.x) >> 5;
  int lane = threadIdx.x & 31;
  int tilesC = C >> 4;
  int rowTiles = (Nrows + 15) >> 4;
  if (wid >= rowTiles * tilesC) return;   // wave-uniform: skips WMMA entirely
  int tM = wid / tilesC, tC = wid % tilesC;
  int half = lane >> 4, l16 = lane & 15;

  int row = tM * 16 + l16;
  int rowc = (row < Nrows) ? row : (Nrows - 1);
  int col = tC * 16 + l16;
  const float* xrow = X + (size_t)rowc * K;
  const float* wcol = W + col;

  v8f acc = {0.f, 0.f, 0.f, 0.f, 0.f, 0.f, 0.f, 0.f};
#pragma unroll
  for (int k0 = 0; k0 < K; k0 += 4) {
    int ka = k0 + half * 2;
    v2f a, b;
    a.x = xrow[ka];
    a.y = xrow[ka + 1];
    b.x = wcol[(size_t)ka * C];
    b.y = wcol[(size_t)(ka + 1) * C];
    acc = __builtin_amdgcn_wmma_f32_16x16x4_f32(
        /*neg_a=*/false, a, /*neg_b=*/false, b,
        /*c_mod=*/(short)0, acc, /*reuse_a=*/false, /*reuse_b=*/false);
  }
#pragma unroll
  for (int r = 0; r < 8; ++r) {
    int orow = tM * 16 + half * 8 + r;
    if (orow < Nrows) Hout[(size_t)orow * C + col] = acc[r];
  }
}

// ======================================================================
// Per-(node,head) attention logits: a_src/a_dst = <h[n,h,:], att>
// ======================================================================
__global__ void gat_att_scores(const float* __restrict__ Hmat,
                               const float* __restrict__ att_src,
                               const float* __restrict__ att_dst,
                               float* __restrict__ asrc,
                               float* __restrict__ adst, int Nn) {
  int idx = blockIdx.x * blockDim.x + threadIdx.x;
  if (idx >= Nn * GAT_H) return;
  int n = idx / GAT_H, hh = idx % GAT_H;
  const float* hp = Hmat + (size_t)n * GAT_HC + hh * GAT_C;
  const float* as = att_src + hh * GAT_C;
  const float* ad = att_dst + hh * GAT_C;
  float s = 0.f, d = 0.f;
#pragma unroll 8
  for (int c = 0; c < GAT_C; ++c) {
    float v = hp[c];
    s += v * as[c];
    d += v * ad[c];
  }
  asrc[idx] = s;
  adst[idx] = d;
}

// ======================================================================
__global__ void gat_zero_u32(unsigned* __restrict__ p, long long n) {
  long long i = (long long)blockIdx.x * blockDim.x + threadIdx.x;
  long long stride = (long long)gridDim.x * blockDim.x;
  for (; i < n; i += stride) p[i] = 0u;   // 0u == encoded -inf, and 0.0f
}

// ======================================================================
// Edge pass 1: segment max of leaky_relu(a_src[s]+a_dst[d]) per head
// ======================================================================
__global__ void gat_edge_max(const float* __restrict__ asrc,
                             const float* __restrict__ adst,
                             const int* __restrict__ eidx,
                             unsigned* __restrict__ mx, int E, int Nn) {
  int i = blockIdx.x * blockDim.x + threadIdx.x;
  int tot = (E + Nn) * GAT_H;
  if (i >= tot) return;
  int ei = i >> 2, hh = i & 3;
  int s, d;
  if (ei < E) { s = eidx[ei]; d = eidx[E + ei]; } else { s = d = ei - E; }
  float v = asrc[s * GAT_H + hh] + adst[d * GAT_H + hh];
  v = (v >= 0.f) ? v : 0.2f * v;
  atomicMax(&mx[d * GAT_H + hh], enc_f32(v));
}

// ======================================================================
// Edge pass 2: denom[d,h] += exp(e - max)
// ======================================================================
__global__ void gat_edge_denom(const float* __restrict__ asrc,
                               const float* __restrict__ adst,
                               const int* __restrict__ eidx,
                               const unsigned* __restrict__ mx,
                               float* __restrict__ denom, int E, int Nn) {
  int i = blockIdx.x * blockDim.x + threadIdx.x;
  int tot = (E + Nn) * GAT_H;
  if (i >= tot) return;
  int ei = i >> 2, hh = i & 3;
  int s, d;
  if (ei < E) { s = eidx[ei]; d = eidx[E + ei]; } else { s = d = ei - E; }
  float v = asrc[s * GAT_H + hh] + adst[d * GAT_H + hh];
  v = (v >= 0.f) ? v : 0.2f * v;
  atomicAdd(&denom[d * GAT_H + hh], __expf(v - dec_f32(mx[d * GAT_H + hh])));
}

// ======================================================================
// Edge pass 3: wave per edge; agg[d, :] += h[s, :] * alpha[head]
// 256 channels = 8 coalesced 128B steps per wave. h is L2-resident (51MB).
// ======================================================================
__global__ void gat_edge_msg(const float* __restrict__ Hmat,
                             const float* __restrict__ asrc,
                             const float* __restrict__ adst,
                             const unsigned* __restrict__ mx,
                             const float* __restrict__ denom,
                             const int* __restrict__ eidx,
                             float* __restrict__ agg, int E, int Nn) {
  int w = (blockIdx.x * blockDim.x + threadIdx.x) >> 5;
  int lane = threadIdx.x & 31;
  if (w >= E + Nn) return;
  int s, d;
  if (w < E) { s = eidx[w]; d = eidx[E + w]; } else { s = d = w - E; }
  float alpha[GAT_H];
#pragma unroll
  for (int hh = 0; hh < GAT_H; ++hh) {
    float v = asrc[s * GAT_H + hh] + adst[d * GAT_H + hh];
    v = (v >= 0.f) ? v : 0.2f * v;
    alpha[hh] = __expf(v - dec_f32(mx[d * GAT_H + hh])) /
                (denom[d * GAT_H + hh] + 1e-16f);
  }
  const float* hs = Hmat + (size_t)s * GAT_HC;
  float* ag = agg + (size_t)d * GAT_HC;
#pragma unroll
  for (int j = 0; j < 8; ++j) {
    int c = j * 32 + lane;          // c/64 == j>>1 for lane<32
    atomicAdd(&ag[c], hs[c] * alpha[j >> 1]);
  }
}

// ======================================================================
// Finalize layer 1: head-mean + bias, LayerNorm, + residual GEMM, ELU -> xh
// One wave per node, 2 channels per lane, shuffle reductions.
// ======================================================================
__global__ void gat_finalize1(const float* __restrict__ agg,
                              const float* __restrict__ b1,
                              const float* __restrict__ gamma,
                              const float* __restrict__ beta,
                              const float* __restrict__ xres,
                              const float* __restrict__ res_b,
                              float* __restrict__ xh, int Nn) {
  int w = (blockIdx.x * blockDim.x + threadIdx.x) >> 5;
  int lane = threadIdx.x & 31;
  if (w >= Nn) return;
  const float* ag = agg + (size_t)w * GAT_HC;
  int c0 = lane, c1 = lane + 32;
  float v0 = 0.25f * (ag[c0] + ag[64 + c0] + ag[128 + c0] + ag[192 + c0]) + b1[c0];
  float v1 = 0.25f * (ag[c1] + ag[64 + c1] + ag[128 + c1] + ag[192 + c1]) + b1[c1];
  float s = v0 + v1;
#pragma unroll
  for (int o = 16; o > 0; o >>= 1) s += __shfl_xor(s, o, 32);
  float mu = s * (1.f / 64.f);
  float q0 = v0 - mu, q1 = v1 - mu;
  float vs = q0 * q0 + q1 * q1;
#pragma unroll
  for (int o = 16; o > 0; o >>= 1) vs += __shfl_xor(vs, o, 32);
  float rstd = rsqrtf(vs * (1.f / 64.f) + LN_EPS_F);
  const float* xr = xres + (size_t)w * GAT_C;
  float y0 = q0 * rstd * gamma[c0] + beta[c0] + xr[c0] + res_b[c0];
  float y1 = q1 * rstd * gamma[c1] + beta[c1] + xr[c1] + res_b[c1];
  y0 = (y0 > 0.f) ? y0 : expm1f(y0);   // ELU, alpha=1
  y1 = (y1 > 0.f) ? y1 : expm1f(y1);
  xh[(size_t)w * GAT_C + c0] = y0;
  xh[(size_t)w * GAT_C + c1] = y1;
}

// ======================================================================
// Finalize layer 2: head-mean + bias, LayerNorm, + xh -> out
// ======================================================================
__global__ void gat_finalize2(const float* __restrict__ agg,
                              const float* __restrict__ b2,
                              const float* __restrict__ gamma,
                              const float* __restrict__ beta,
                              const float* __restrict__ xh,
                              float* __restrict__ out, int Nn) {
  int w = (blockIdx.x * blockDim.x + threadIdx.x) >> 5;
  int lane = threadIdx.x & 31;
  if (w >= Nn) return;
  const float* ag = agg + (size_t)w * GAT_HC;
  int c0 = lane, c1 = lane + 32;
  float v0 = 0.25f * (ag[c0] + ag[64 + c0] + ag[128 + c0] + ag[192 + c0]) + b2[c0];
  float v1 = 0.25f * (ag[c1] + ag[64 + c1] + ag[128 + c1] + ag[192 + c1]) + b2[c1];
  float s = v0 + v1;
#pragma unroll
  for (int o = 16; o > 0; o >>= 1) s += __shfl_xor(s, o, 32);
  float mu = s * (1.f / 64.f);
  float q0 = v0 - mu, q1 = v1 - mu;
  float vs = q0 * q0 + q1 * q1;
#pragma unroll
  for (int o = 16; o > 0; o >>= 1) vs += __shfl_xor(vs, o, 32);
  float rstd = rsqrtf(vs * (1.f / 64.f) + LN_EPS_F);
  out[(size_t)w * GAT_C + c0] = q0 * rstd * gamma[c0] + beta[c0] + xh[(size_t)w * GAT_C + c0];
  out[(size_t)w * GAT_C + c1] = q1 * rstd * gamma[c1] + beta[c1] + xh[(size_t)w * GAT_C + c1];
}

// ======================================================================
extern "C" void kernel_launch(void* const* d_in, const int* in_sizes, int n_in,
                              void* d_out, int out_size, void* d_ws, size_t ws_size,
                              hipStream_t stream) {
  const float* x        = (const float*)d_in[0];
  const int*   eidx     = (const int*)d_in[1];
  const float* W1       = (const float*)d_in[2];
  const float* att_src1 = (const float*)d_in[3];
  const float* att_dst1 = (const float*)d_in[4];
  const float* b1       = (const float*)d_in[5];
  const float* gamma1   = (const float*)d_in[6];
  const float* beta1    = (const float*)d_in[7];
  const float* res_w    = (const float*)d_in[8];
  const float* res_b    = (const float*)d_in[9];
  const float* W2       = (const float*)d_in[10];
  const float* att_src2 = (const float*)d_in[11];
  const float* att_dst2 = (const float*)d_in[12];
  const float* b2       = (const float*)d_in[13];
  const float* gamma2   = (const float*)d_in[14];
  const float* beta2    = (const float*)d_in[15];

  const int N = in_sizes[0] / 128;
  const int E = in_sizes[1] / 2;
  float* out = (float*)d_out;

  // Workspace carve-out (agg|mx|denom kept contiguous for one-shot re-zero).
  char* ws = (char*)d_ws;
  size_t off = 0;
  auto carve = [&](size_t bytes) -> void* {
    void* p = ws + off;
    off += (bytes + 255) & ~(size_t)255;
    return p;
  };
  float*    h     = (float*)   carve((size_t)N * GAT_HC * 4);  // [N,256]
  float*    xres  = (float*)   carve((size_t)N * GAT_C * 4);   // [N,64]
  float*    xh    = (float*)   carve((size_t)N * GAT_C * 4);   // [N,64]
  float*    asrc  = (float*)   carve((size_t)N * GAT_H * 4);
  float*    adst  = (float*)   carve((size_t)N * GAT_H * 4);
  float*    agg   = (float*)   carve((size_t)N * GAT_HC * 4);  // [N,256]
  unsigned* mx    = (unsigned*)carve((size_t)N * GAT_H * 4);
  float*    denom = (float*)   carve((size_t)N * GAT_H * 4);

  const int BLK = 256;
  const int rowTiles = (N + 15) / 16;
  const long long zeroN = (long long)N * (GAT_HC + 2 * GAT_H); // agg+mx+denom

  // ---------------- Layer 1 ----------------
  {
    int waves = rowTiles * (GAT_HC / 16);
    gat_gemm_wmma<128><<<dim3((waves * 32 + BLK - 1) / BLK), BLK, 0, stream>>>(
        x, W1, h, N, GAT_HC);
  }
  {
    int waves = rowTiles * (GAT_C / 16);
    gat_gemm_wmma<128><<<dim3((waves * 32 + BLK - 1) / BLK), BLK, 0, stream>>>(
        x, res_w, xres, N, GAT_C);
  }
  gat_att_scores<<<dim3((N * GAT_H + BLK - 1) / BLK), BLK, 0, stream>>>(
      h, att_src1, att_dst1, asrc, adst, N);
  gat_zero_u32<<<dim3(2048), BLK, 0, stream>>>((unsigned*)agg, zeroN);

  {
    int tot = (E + N) * GAT_H;
    gat_edge_max<<<dim3((tot + BLK - 1) / BLK), BLK, 0, stream>>>(
        asrc, adst, eidx, mx, E, N);
    gat_edge_denom<<<dim3((tot + BLK - 1) / BLK), BLK, 0, stream>>>(
        asrc, adst, eidx, mx, denom, E, N);
    long long thr = (long long)(E + N) * 32;
    gat_edge_msg<<<dim3((unsigned)((thr + BLK - 1) / BLK)), BLK, 0, stream>>>(
        h, asrc, adst, mx, denom, eidx, agg, E, N);
  }
  gat_finalize1<<<dim3(((long long)N * 32 + BLK - 1) / BLK), BLK, 0, stream>>>(
      agg, b1, gamma1, beta1, xres, res_b, xh, N);

  // ---------------- Layer 2 ----------------
  {
    int waves = rowTiles * (GAT_HC / 16);
    gat_gemm_wmma<64><<<dim3((waves * 32 + BLK - 1) / BLK), BLK, 0, stream>>>(
        xh, W2, h, N, GAT_HC);
  }
  gat_att_scores<<<dim3((N * GAT_H + BLK - 1) / BLK), BLK, 0, stream>>>(
      h, att_src2, att_dst2, asrc, adst, N);
  gat_zero_u32<<<dim3(2048), BLK, 0, stream>>>((unsigned*)agg, zeroN);

  {
    int tot = (E + N) * GAT_H;
    gat_edge_max<<<dim3((tot + BLK - 1) / BLK), BLK, 0, stream>>>(
        asrc, adst, eidx, mx, E, N);
    gat_edge_denom<<<dim3((tot + BLK - 1) / BLK), BLK, 0, stream>>>(
        asrc, adst, eidx, mx, denom, E, N);
    long long thr = (long long)(E + N) * 32;
    gat_edge_msg<<<dim3((unsigned)((thr + BLK - 1) / BLK)), BLK, 0, stream>>>(
        h, asrc, adst, mx, denom, eidx, agg, E, N);
  }
  gat_finalize2<<<dim3(((long long)N * 32 + BLK - 1) / BLK), BLK, 0, stream>>>(
      agg, b2, gamma2, beta2, xh, out, N);
}